// AttentionLayerDecoder_6270652252637
// MI455X (gfx1250) — compile-verified
//
#include <hip/hip_runtime.h>
#include <hip/hip_bf16.h>
#include <math.h>

#define NN    100000
#define NB    128
#define NH    8
#define DCC   256
#define DE4   128   // DE+4 = x feature width
#define DV    64
#define DEOUT 124
#define SPLIT 8     // chunks per graph in pass 2

typedef __attribute__((ext_vector_type(2))) float v2f;
typedef __attribute__((ext_vector_type(8))) float v8f;

// ---------------- monotone float<->uint encoding for atomic max ----------------
__device__ __forceinline__ unsigned encf(float f) {
    unsigned b = __float_as_uint(f);
    return (b & 0x80000000u) ? ~b : (b | 0x80000000u);
}
__device__ __forceinline__ float decf(unsigned k) {
    return (k & 0x80000000u) ? __uint_as_float(k & 0x7FFFFFFFu) : __uint_as_float(~k);
}

// ---------------- f32 WMMA helpers (V_WMMA_F32_16X16X4_F32) ----------------
// A(m,k) = A[m*asM + k*asK], B(k,n) = B[k*bsK + n*bsN]
// ISA layout: A 16x4 -> lane l16 rows M=l16; lane<16 holds K=0,1 ; lane>=16 K=2,3.
__device__ __forceinline__ v8f wmma_step(const float* A, int asM, int asK,
                                         const float* B, int bsK, int bsN,
                                         int m0, int n0, int k0,
                                         int l16, int koff, v8f acc) {
    v2f a, b;
    a[0] = A[(size_t)(m0 + l16) * asM + (size_t)(k0 + koff + 0) * asK];
    a[1] = A[(size_t)(m0 + l16) * asM + (size_t)(k0 + koff + 1) * asK];
    b[0] = B[(size_t)(k0 + koff + 0) * bsK + (size_t)(n0 + l16) * bsN];
    b[1] = B[(size_t)(k0 + koff + 1) * bsK + (size_t)(n0 + l16) * bsN];
    return __builtin_amdgcn_wmma_f32_16x16x4_f32(false, a, false, b, (short)0, acc, false, false);
}

// C/D 16x16: VGPR r holds row m0+r (lanes 0-15) and m0+r+8 (lanes 16-31)
__device__ __forceinline__ void store_tile(float* D, int ld, int m0, int n0, v8f acc, float scale) {
    int lane = threadIdx.x & 31, half = lane >> 4, l16 = lane & 15;
#pragma unroll
    for (int r = 0; r < 8; ++r)
        D[(size_t)(m0 + r + 8 * half) * ld + n0 + l16] = acc[r] * scale;
}

// ---------------- kernel 0: init umax / zero den+s ----------------
__global__ void k_init(unsigned* umax, float* denz) {
    int i = blockIdx.x * blockDim.x + threadIdx.x;
    if (i < NH * NB) umax[i] = encf(-INFINITY);
    if (i < NH * NB + NH * NB * DE4) denz[i] = 0.0f;  // den (1024) followed by s (131072), contiguous
}

// ---------------- kernel 1: query[h] = context @ Wq[h]  (128x256 @ 256x64) ----------------
__global__ void __launch_bounds__(32) k_query(const float* __restrict__ context,
                                              const float* __restrict__ Wq,
                                              float* __restrict__ query) {
    int bid = blockIdx.x;
    int h = bid >> 5, rt = (bid >> 2) & 7, ct = bid & 3;
    int lane = threadIdx.x & 31, l16 = lane & 15, koff = (lane >> 4) * 2;
    v8f acc = {0, 0, 0, 0, 0, 0, 0, 0};
    const float* Bh = Wq + (size_t)h * DCC * DV;
    for (int k0 = 0; k0 < DCC; k0 += 4)
        acc = wmma_step(context, DCC, 1, Bh, DV, 1, rt * 16, ct * 16, k0, l16, koff, acc);
    store_tile(query + (size_t)h * NB * DV, DV, rt * 16, ct * 16, acc, 1.0f);
}

// ---------------- kernel 2: wu[h][b][d] = (1/8) * sum_v query[h][b][v] * Wk[h][d][v] ----------------
__global__ void __launch_bounds__(32) k_wu(const float* __restrict__ query,
                                           const float* __restrict__ Wk,
                                           float* __restrict__ wu) {
    int bid = blockIdx.x;
    int h = bid >> 6, rt = (bid >> 3) & 7, ct = bid & 7;
    int lane = threadIdx.x & 31, l16 = lane & 15, koff = (lane >> 4) * 2;
    v8f acc = {0, 0, 0, 0, 0, 0, 0, 0};
    const float* Ah = query + (size_t)h * NB * DV;   // A(m=b, k=v), asM=DV
    const float* Bh = Wk + (size_t)h * DE4 * DV;     // B(k=v, n=d) = Wk[h][d][v] -> bsK=1, bsN=DV
    for (int k0 = 0; k0 < DV; k0 += 4)
        acc = wmma_step(Ah, DV, 1, Bh, 1, DV, rt * 16, ct * 16, k0, l16, koff, acc);
    store_tile(wu + (size_t)h * NB * DE4, DE4, rt * 16, ct * 16, acc, 0.125f);  // 1/sqrt(DV)
}

// ---------------- kernel 3: pass 1 over x: u[h,n] = x_n . wu[h, batch[n]], seg-max ----------------
__global__ void __launch_bounds__(256) k_pass1(const float4* __restrict__ x4,
                                               const int* __restrict__ batch,
                                               const float* __restrict__ wu,
                                               float* __restrict__ u,
                                               unsigned* __restrict__ umax) {
    int n = blockIdx.x * blockDim.x + threadIdx.x;
    bool valid = n < NN;
    int nc = valid ? n : (NN - 1);
    int b = batch[nc];
    float acc[NH];
#pragma unroll
    for (int h = 0; h < NH; ++h) acc[h] = 0.0f;
    const float4* xr = x4 + (size_t)nc * (DE4 / 4);
    const float4* wu4 = (const float4*)wu;
    for (int d4 = 0; d4 < DE4 / 4; ++d4) {
        float4 xv = xr[d4];
#pragma unroll
        for (int h = 0; h < NH; ++h) {
            float4 wv = wu4[(size_t)(h * NB + b) * (DE4 / 4) + d4];
            acc[h] += xv.x * wv.x + xv.y * wv.y + xv.z * wv.z + xv.w * wv.w;
        }
    }
    if (valid) {
#pragma unroll
        for (int h = 0; h < NH; ++h) u[(size_t)h * NN + n] = acc[h];
    } else {
#pragma unroll
        for (int h = 0; h < NH; ++h) acc[h] = -INFINITY;
    }
    // wave-level max reduction when all 32 lanes share a graph (common: ~781 nodes/graph)
    int b0 = __shfl(b, 0, 32);
    if (__all(b == b0)) {
#pragma unroll
        for (int h = 0; h < NH; ++h) {
            float m = acc[h];
            for (int off = 16; off > 0; off >>= 1) m = fmaxf(m, __shfl_xor(m, off, 32));
            if ((threadIdx.x & 31) == 0) atomicMax(&umax[h * NB + b0], encf(m));
        }
    } else if (valid) {
#pragma unroll
        for (int h = 0; h < NH; ++h) atomicMax(&umax[h * NB + b], encf(acc[h]));
    }
}

__device__ __forceinline__ int lower_bound_dev(const int* a, int n, int v) {
    int lo = 0, hi = n;
    while (lo < hi) { int mid = (lo + hi) >> 1; if (a[mid] < v) lo = mid + 1; else hi = mid; }
    return lo;
}

// ---------------- kernel 4: pass 2: e = exp(u - max); s[h,b,:] += e*x_n; den += e ----------------
__global__ void __launch_bounds__(256) k_seg(const float4* __restrict__ x4,
                                             const int* __restrict__ batch,
                                             const float* __restrict__ u,
                                             const unsigned* __restrict__ umax,
                                             float* __restrict__ s,
                                             float* __restrict__ den) {
    __shared__ float xs[64 * DE4];     // 32 KB node tile
    __shared__ float es[NH * 64];      // e values for tile
    __shared__ float mxs[NH];
    __shared__ int seg[2];

    int b = blockIdx.x >> 3, c = blockIdx.x & (SPLIT - 1);
    if (threadIdx.x == 0) {
        seg[0] = lower_bound_dev(batch, NN, b);
        seg[1] = lower_bound_dev(batch, NN, b + 1);
    }
    if (threadIdx.x < NH) mxs[threadIdx.x] = decf(umax[threadIdx.x * NB + b]);
    __syncthreads();
    int start = seg[0], end = seg[1], len = end - start;
    int cstart = start + (int)(((long)len * c) / SPLIT);
    int cend   = start + (int)(((long)len * (c + 1)) / SPLIT);

    int h = threadIdx.x >> 5, dg = threadIdx.x & 31;
    float ax = 0.f, ay = 0.f, az = 0.f, aw = 0.f, dacc = 0.f;

    for (int t0 = cstart; t0 < cend; t0 += 64) {
        __syncthreads();  // protect previous tile reads
        // stage 64 x-rows (2048 float4) into LDS
        for (int i = threadIdx.x; i < 64 * (DE4 / 4); i += 256) {
            int node = t0 + (i >> 5);
            float4 v = {0.f, 0.f, 0.f, 0.f};
            if (node < cend) v = x4[(size_t)node * (DE4 / 4) + (i & 31)];
            ((float4*)xs)[i] = v;
        }
        // e values: 8 heads x 64 nodes
        for (int j = threadIdx.x; j < NH * 64; j += 256) {
            int hh = j >> 6, i = j & 63, node = t0 + i;
            float e = 0.f;
            if (node < cend) e = __expf(u[(size_t)hh * NN + node] - mxs[hh]);
            es[j] = e;
        }
        __syncthreads();
        // thread (h, dg) accumulates s[h, b, 4*dg .. 4*dg+3]
#pragma unroll 4
        for (int i = 0; i < 64; ++i) {
            float ev = es[h * 64 + i];
            float4 xv = ((const float4*)xs)[i * (DE4 / 4) + dg];
            ax += ev * xv.x; ay += ev * xv.y; az += ev * xv.z; aw += ev * xv.w;
        }
        if (dg == 0)
            for (int i = 0; i < 64; ++i) dacc += es[h * 64 + i];
    }
    float* sp = s + ((size_t)(h * NB + b) * DE4) + dg * 4;
    atomicAdd(sp + 0, ax); atomicAdd(sp + 1, ay);
    atomicAdd(sp + 2, az); atomicAdd(sp + 3, aw);
    if (dg == 0) atomicAdd(&den[h * NB + b], dacc);
}

// ---------------- kernel 5: T[b,v] = qc*sum_h query + sum_h (s[h,b,:]/den) @ Wv[h] ----------------
__global__ void __launch_bounds__(32) k_T(const float* __restrict__ s,
                                          const float* __restrict__ den,
                                          const float* __restrict__ Wv,
                                          const float* __restrict__ query,
                                          const float* __restrict__ qcp,
                                          float* __restrict__ T) {
    int rt = blockIdx.x >> 2, ct = blockIdx.x & 3;
    int lane = threadIdx.x & 31, l16 = lane & 15, half = lane >> 4, koff = half * 2;
    int m0 = rt * 16, n0 = ct * 16;
    v8f acc = {0, 0, 0, 0, 0, 0, 0, 0};
    for (int h = 0; h < NH; ++h) {
        float invd = 1.0f / (den[h * NB + m0 + l16] + 1e-16f);   // per-row scale (a = e/(sum+eps))
        const float* Ah = s + (size_t)h * NB * DE4;
        const float* Bh = Wv + (size_t)h * DE4 * DV;
        for (int k0 = 0; k0 < DE4; k0 += 4) {
            v2f a, bb;
            a[0] = Ah[(size_t)(m0 + l16) * DE4 + k0 + koff + 0] * invd;
            a[1] = Ah[(size_t)(m0 + l16) * DE4 + k0 + koff + 1] * invd;
            bb[0] = Bh[(size_t)(k0 + koff + 0) * DV + n0 + l16];
            bb[1] = Bh[(size_t)(k0 + koff + 1) * DV + n0 + l16];
            acc = __builtin_amdgcn_wmma_f32_16x16x4_f32(false, a, false, bb, (short)0, acc, false, false);
        }
    }
    float qcv = qcp[0];
#pragma unroll
    for (int r = 0; r < 8; ++r) {
        int m = m0 + r + 8 * half, n = n0 + l16;
        float qs = 0.f;
#pragma unroll
        for (int h = 0; h < NH; ++h) qs += query[(size_t)h * NB * DV + m * DV + n];
        T[m * DV + n] = acc[r] + qcv * qs;
    }
}

// ---------------- kernel 6: out = T @ Wf  (128x64 @ 64x124) ----------------
__global__ void __launch_bounds__(32) k_out(const float* __restrict__ T,
                                            const float* __restrict__ Wf,
                                            float* __restrict__ out) {
    int rt = blockIdx.x >> 3, ct = blockIdx.x & 7;
    int lane = threadIdx.x & 31, l16 = lane & 15, half = lane >> 4, koff = half * 2;
    int m0 = rt * 16, n0 = ct * 16;
    bool ok = (n0 + l16) < DEOUT;
    v8f acc = {0, 0, 0, 0, 0, 0, 0, 0};
    for (int k0 = 0; k0 < DV; k0 += 4) {
        v2f a, bb;
        a[0] = T[(size_t)(m0 + l16) * DV + k0 + koff + 0];
        a[1] = T[(size_t)(m0 + l16) * DV + k0 + koff + 1];
        bb[0] = ok ? Wf[(size_t)(k0 + koff + 0) * DEOUT + n0 + l16] : 0.f;
        bb[1] = ok ? Wf[(size_t)(k0 + koff + 1) * DEOUT + n0 + l16] : 0.f;
        acc = __builtin_amdgcn_wmma_f32_16x16x4_f32(false, a, false, bb, (short)0, acc, false, false);
    }
#pragma unroll
    for (int r = 0; r < 8; ++r) {
        int m = m0 + r + 8 * half, n = n0 + l16;
        if (n < DEOUT) out[(size_t)m * DEOUT + n] = acc[r];
    }
}

extern "C" void kernel_launch(void* const* d_in, const int* in_sizes, int n_in,
                              void* d_out, int out_size, void* d_ws, size_t ws_size,
                              hipStream_t stream) {
    (void)in_sizes; (void)n_in; (void)out_size; (void)ws_size;
    const float* x       = (const float*)d_in[0];
    const int*   batch   = (const int*)d_in[2];   // edge_index (d_in[1]) unused by reference
    const float* context = (const float*)d_in[3];
    const float* Wq      = (const float*)d_in[4];
    const float* Wk      = (const float*)d_in[5];
    const float* Wv      = (const float*)d_in[6];
    const float* qc      = (const float*)d_in[7];
    const float* Wf      = (const float*)d_in[8];
    float* out = (float*)d_out;

    // workspace layout (floats)
    float*    query = (float*)d_ws;                  // 8*128*64   = 65536
    float*    wu    = query + NH * NB * DV;          // 8*128*128  = 131072
    float*    u     = wu + NH * NB * DE4;            // 8*100000   = 800000
    unsigned* umax  = (unsigned*)(u + NH * NN);      // 1024
    float*    den   = (float*)(umax + NH * NB);      // 1024   (den,s contiguous for zeroing)
    float*    s     = den + NH * NB;                 // 131072
    float*    T     = s + NH * NB * DE4;             // 8192
    // total ~4.55 MB

    int initN = NH * NB + NH * NB * DE4;  // covers umax + (den,s)
    k_init <<<(initN + 255) / 256, 256, 0, stream>>>(umax, den);
    k_query<<<NH * 8 * 4, 32, 0, stream>>>(context, Wq, query);
    k_wu   <<<NH * 8 * 8, 32, 0, stream>>>(query, Wk, wu);
    k_pass1<<<(NN + 255) / 256, 256, 0, stream>>>((const float4*)x, batch, wu, u, umax);
    k_seg  <<<NB * SPLIT, 256, 0, stream>>>((const float4*)x, batch, u, umax, s, den);
    k_T    <<<8 * 4, 32, 0, stream>>>(s, den, Wv, query, qc, T);
    k_out  <<<8 * 8, 32, 0, stream>>>(T, Wf, out);
}